// GATLayer_40510131535939
// MI455X (gfx1250) — compile-verified
//
#include <hip/hip_runtime.h>
#include <hip/hip_bf16.h>
#include <math.h>

#define N_NODES 100000
#define N_EDGES 1600000
#define DIM     32
#define SLOPE   0.01f

typedef float v2f __attribute__((ext_vector_type(2)));
typedef float v8f __attribute__((ext_vector_type(8)));

// ---------------------------------------------------------------------------
// Kernel 1: init accumulators. d_out is poisoned by the harness and all later
// writes are atomic accumulations, so everything must be re-zeroed each call.
// ---------------------------------------------------------------------------
__global__ void gat_init(float* __restrict__ out,
                         float* __restrict__ m,
                         float* __restrict__ denom) {
    int i = blockIdx.x * blockDim.x + threadIdx.x;
    if (i < N_NODES * DIM) out[i] = 0.0f;
    if (i < N_NODES) {
        m[i]     = -INFINITY;
        denom[i] = 0.0f;
    }
}

// ---------------------------------------------------------------------------
// Kernel 2: dual GEMV  s_src = h @ w[:32],  s_dst = h @ w[32:]
// via V_WMMA_F32_16X16X4_F32. One wave32 handles a 16-node tile; B-matrix
// column 0 carries w_src, column 1 carries w_dst, so both score vectors come
// out of a single accumulation chain of 8 WMMAs (K = 32 = 8 x 4).
//
// f32 16x16x4 VGPR layout (ISA 7.12.2):
//   A (16x4):  lane L, reg r  ->  M = L&15,      K = 2*(L>>4) + r
//   B (4x16):  lane L, reg r  ->  N = L&15,      K = 2*(L>>4) + r
//   D (16x16): reg r, lanes 0-15: (M=r,  N=lane); lanes 16-31: (M=8+r, N=lane-16)
//
// B is built branch-free: every lane issues an unconditional (always-valid)
// load of w[(N&1)*32 + kb] and scales it by a lane mask, so no EXEC-mask
// manipulation happens anywhere near the WMMAs (EXEC stays all-ones).
// ---------------------------------------------------------------------------
__global__ void gat_scores_wmma(const float* __restrict__ h,
                                const float* __restrict__ w,
                                float* __restrict__ s_src,
                                float* __restrict__ s_dst) {
    const int lane = threadIdx.x & 31;
    const int wave = threadIdx.x >> 5;
    const int tile = blockIdx.x * (blockDim.x >> 5) + wave;
    const int n_tiles = N_NODES / 16;               // 6250, exact
    if (tile >= n_tiles) return;                    // wave-uniform exit

    const int base = tile * 16;
    const int M    = lane & 15;                     // row for A, col for B/D
    const int half = lane >> 4;                     // K-group select
    const float* __restrict__ hrow = h + (size_t)(base + M) * DIM;
    // Lane's B source row: col 0 -> w_src (w[0:32]), col 1 -> w_dst (w[32:64]).
    // Cols >= 2 load a valid address but are zeroed by the mask.
    const float* __restrict__ wp = w + (M & 1) * DIM;
    const float  bmask = (M < 2) ? 1.0f : 0.0f;

    v8f acc = {0.f, 0.f, 0.f, 0.f, 0.f, 0.f, 0.f, 0.f};

#pragma unroll
    for (int kk = 0; kk < 8; ++kk) {
        const int kb = kk * 4 + 2 * half;           // this lane's K pair base
        v2f a = *(const v2f*)(hrow + kb);           // global_load_b64
        v2f bw = *(const v2f*)(wp + kb);            // global_load_b64 (L0-hot)
        v2f b;
        b.x = bw.x * bmask;
        b.y = bw.y * bmask;
        // (neg_a, A, neg_b, B, c_mod, C, reuse_a, reuse_b)
        acc = __builtin_amdgcn_wmma_f32_16x16x4_f32(
            false, a, false, b, (short)0, acc, false, false);
    }

    // Column 0 of D -> s_src, column 1 -> s_dst (divergence after WMMAs is OK).
    if (M < 2) {
        float* __restrict__ dstp = (M == 0) ? s_src : s_dst;
        const int row0 = base + 8 * half;
#pragma unroll
        for (int r = 0; r < 8; ++r) dstp[row0 + r] = acc[r];
    }
}

// ---------------------------------------------------------------------------
// Edge-phase helpers
// ---------------------------------------------------------------------------
__device__ __forceinline__ float leaky(float e) {
    return e >= 0.0f ? e : SLOPE * e;
}

// Order-independent float atomic max via integer CAS (m starts at -inf).
__device__ __forceinline__ void atomicMaxF(float* addr, float val) {
    unsigned int* ua  = (unsigned int*)addr;
    unsigned int  old = *ua;
    while (__uint_as_float(old) < val) {
        unsigned int assumed = old;
        old = atomicCAS(ua, assumed, __float_as_uint(val));
        if (old == assumed) break;
    }
}

// Kernel 3: m[dst] = segment_max(leaky(e))
__global__ void gat_edge_max(const int* __restrict__ src,
                             const int* __restrict__ dst,
                             const float* __restrict__ s_src,
                             const float* __restrict__ s_dst,
                             float* __restrict__ m) {
    int i = blockIdx.x * blockDim.x + threadIdx.x;
    if (i >= N_EDGES) return;
    const int s = src[i];
    const int d = dst[i];
    const float e = leaky(s_src[s] + s_dst[d]);
    atomicMaxF(&m[d], e);
}

// Kernel 4: denom[dst] = segment_sum(exp(e - m[dst]))
__global__ void gat_edge_sum(const int* __restrict__ src,
                             const int* __restrict__ dst,
                             const float* __restrict__ s_src,
                             const float* __restrict__ s_dst,
                             const float* __restrict__ m,
                             float* __restrict__ denom) {
    int i = blockIdx.x * blockDim.x + threadIdx.x;
    if (i >= N_EDGES) return;
    const int s = src[i];
    const int d = dst[i];
    const float e  = leaky(s_src[s] + s_dst[d]);
    const float ex = __expf(e - m[d]);
    atomicAdd(&denom[d], ex);
}

// Kernel 5: out[dst] += alpha * h[src]. One wave per edge, one lane per
// feature dim -> each edge issues a fully coalesced 128B read of h[src] and
// 32 consecutive-address f32 atomic adds into out[dst].
__global__ void gat_aggregate(const int* __restrict__ src,
                              const int* __restrict__ dst,
                              const float* __restrict__ s_src,
                              const float* __restrict__ s_dst,
                              const float* __restrict__ m,
                              const float* __restrict__ denom,
                              const float* __restrict__ h,
                              float* __restrict__ out) {
    const int lane  = threadIdx.x & 31;
    const int eidx  = (blockIdx.x * blockDim.x + threadIdx.x) >> 5;
    if (eidx >= N_EDGES) return;
    const int s = src[eidx];
    const int d = dst[eidx];
    const float e     = leaky(s_src[s] + s_dst[d]);
    const float alpha = __expf(e - m[d]) / denom[d];
    const float v = h[(size_t)s * DIM + lane] * alpha;
    atomicAdd(&out[(size_t)d * DIM + lane], v);
}

// ---------------------------------------------------------------------------
// Launch
// ---------------------------------------------------------------------------
extern "C" void kernel_launch(void* const* d_in, const int* in_sizes, int n_in,
                              void* d_out, int out_size, void* d_ws, size_t ws_size,
                              hipStream_t stream) {
    const float* h      = (const float*)d_in[0];   // [N_NODES * DIM]
    const float* attn_w = (const float*)d_in[1];   // [2 * DIM]
    const int*   src    = (const int*)d_in[2];     // [N_EDGES]
    const int*   dst    = (const int*)d_in[3];     // [N_EDGES]
    float*       out    = (float*)d_out;           // [N_NODES * DIM]

    // Workspace: 4 * N_NODES floats = 1.6 MB
    float* s_src = (float*)d_ws;
    float* s_dst = s_src + N_NODES;
    float* m     = s_dst + N_NODES;
    float* denom = m + N_NODES;

    {   // 1) init out / m / denom
        int total = N_NODES * DIM;
        gat_init<<<(total + 255) / 256, 256, 0, stream>>>(out, m, denom);
    }
    {   // 2) WMMA dual-GEMV scores: 6250 tiles, 8 waves (tiles) per block
        int n_tiles = N_NODES / 16;
        int tiles_per_block = 256 / 32;
        int grid = (n_tiles + tiles_per_block - 1) / tiles_per_block;
        gat_scores_wmma<<<grid, 256, 0, stream>>>(h, attn_w, s_src, s_dst);
    }
    {   // 3) segment max
        gat_edge_max<<<(N_EDGES + 255) / 256, 256, 0, stream>>>(src, dst, s_src, s_dst, m);
    }
    {   // 4) segment sum of exp
        gat_edge_sum<<<(N_EDGES + 255) / 256, 256, 0, stream>>>(src, dst, s_src, s_dst, m, denom);
    }
    {   // 5) weighted scatter-add, one wave per edge
        long long threads = (long long)N_EDGES * 32;
        int grid = (int)((threads + 255) / 256);
        gat_aggregate<<<grid, 256, 0, stream>>>(src, dst, s_src, s_dst, m, denom, h, out);
    }
}